// GAT_31971736552025
// MI455X (gfx1250) — compile-verified
//
#include <hip/hip_runtime.h>
#include <math.h>

// ---------------- GAT forward for MI455X (gfx1250) ----------------
// h = x@W via V_WMMA_F32_16X16X4_F32 (exact f32), then edge softmax +
// scatter-aggregate using CDNA5 global f32 atomics. Memory-bound design:
// ~0.5 GB total traffic -> ~25us floor at 23.3 TB/s.

typedef __attribute__((ext_vector_type(2))) float v2f;
typedef __attribute__((ext_vector_type(8))) float v8f;

#define F_IN      128
#define HEADS     4
#define COUT      16
#define HC        64         // HEADS * COUT
#define NEG_SLOPE 0.2f

// float atomic max via sign-split int/uint atomics (valid with -inf init)
__device__ __forceinline__ void atomicMaxF32(float* addr, float v) {
  if (v >= 0.0f)
    atomicMax(reinterpret_cast<int*>(addr), __float_as_int(v));
  else
    atomicMin(reinterpret_cast<unsigned int*>(addr), __float_as_uint(v));
}

// ---- init: out = bias (scatter target), m = -inf, s = 0 ----
__global__ void k_init(float* __restrict__ out, const float* __restrict__ bias,
                       float* __restrict__ m, float* __restrict__ s, int N) {
  int idx = blockIdx.x * blockDim.x + threadIdx.x;
  if (idx < N * HC) out[idx] = bias[idx & (HC - 1)];
  if (idx < N * HEADS) { m[idx] = -INFINITY; s[idx] = 0.0f; }
}

// ---- GEMM: h[N,64] = x[N,128] @ W[128,64] via WMMA f32 16x16x4 ----
// Block = 128 threads (4 waves). Block owns 16 rows; wave w owns head w
// (cols 16w..16w+15). 32 K-steps of 4.
__global__ void k_gemm_wmma(const float* __restrict__ x, const float* __restrict__ W,
                            float* __restrict__ hbuf, int N) {
  const int row0 = blockIdx.x * 16;
  const int wave = threadIdx.x >> 5;         // head / N-tile
  const int lane = threadIdx.x & 31;
  const int mrow = lane & 15;
  const int khalf = (lane >> 4) << 1;        // lanes 16-31 carry K+2,K+3
  const int ncol = wave * COUT + (lane & 15);

  int arow = row0 + mrow;
  if (arow >= N) arow = N - 1;               // clamp reads; EXEC stays all-1s
  const float* __restrict__ xrow = x + (size_t)arow * F_IN;

  v8f acc = {};
#pragma unroll
  for (int kk = 0; kk < F_IN; kk += 4) {
    const int k0 = kk + khalf;
    v2f a, b;
    a.x = xrow[k0];
    a.y = xrow[k0 + 1];
    b.x = W[(size_t)k0 * HC + ncol];
    b.y = W[(size_t)(k0 + 1) * HC + ncol];
    acc = __builtin_amdgcn_wmma_f32_16x16x4_f32(
        /*neg_a=*/false, a, /*neg_b=*/false, b,
        /*c_mod=*/(short)0, acc, /*reuse_a=*/false, /*reuse_b=*/false);
  }

  // C/D layout: VGPR r -> row r (lanes 0-15) / row r+8 (lanes 16-31)
  const int mbase = row0 + ((lane >> 4) << 3);
#pragma unroll
  for (int r = 0; r < 8; ++r) {
    const int mr = mbase + r;
    if (mr < N) hbuf[(size_t)mr * HC + ncol] = acc[r];
  }
}

// ---- attention coefficients: a_src[n,h], a_dst[n,h] ----
__global__ void k_attn(const float* __restrict__ hbuf,
                       const float* __restrict__ att_src,
                       const float* __restrict__ att_dst,
                       float* __restrict__ a_src, float* __restrict__ a_dst, int N) {
  int idx = blockIdx.x * blockDim.x + threadIdx.x;
  if (idx >= N * HEADS) return;
  const int n = idx >> 2, hd = idx & 3;
  const float* __restrict__ hp = hbuf + (size_t)n * HC + hd * COUT;
  const float* __restrict__ as = att_src + hd * COUT;
  const float* __restrict__ ad = att_dst + hd * COUT;
  float ssum = 0.0f, dsum = 0.0f;
#pragma unroll
  for (int c = 0; c < COUT; ++c) {
    const float v = hp[c];
    ssum += v * as[c];
    dsum += v * ad[c];
  }
  a_src[idx] = ssum;
  a_dst[idx] = dsum;
}

__device__ __forceinline__ void edge_pair(const long long* __restrict__ ei,
                                          long long e, long long E,
                                          long long& src, long long& dst) {
  if (e < E) { src = ei[e]; dst = ei[E + e]; }
  else       { src = dst = e - E; }           // self-loop
}

// ---- pass 1: segment max of leaky-relu logits over dst ----
__global__ void k_edge_max(const long long* __restrict__ ei,
                           const float* __restrict__ a_src,
                           const float* __restrict__ a_dst,
                           float* __restrict__ m, long long E, long long ET) {
  long long idx = (long long)blockIdx.x * blockDim.x + threadIdx.x;
  if (idx >= ET * HEADS) return;
  const long long e = idx >> 2;
  const int hd = (int)(idx & 3);
  long long src, dst;
  edge_pair(ei, e, E, src, dst);
  float v = a_src[src * HEADS + hd] + a_dst[dst * HEADS + hd];
  v = (v > 0.0f) ? v : NEG_SLOPE * v;
  atomicMaxF32(&m[dst * HEADS + hd], v);
}

// ---- pass 2: segment sum of exp(e - m[dst]) ----
__global__ void k_edge_sum(const long long* __restrict__ ei,
                           const float* __restrict__ a_src,
                           const float* __restrict__ a_dst,
                           const float* __restrict__ m, float* __restrict__ s,
                           long long E, long long ET) {
  long long idx = (long long)blockIdx.x * blockDim.x + threadIdx.x;
  if (idx >= ET * HEADS) return;
  const long long e = idx >> 2;
  const int hd = (int)(idx & 3);
  long long src, dst;
  edge_pair(ei, e, E, src, dst);
  float v = a_src[src * HEADS + hd] + a_dst[dst * HEADS + hd];
  v = (v > 0.0f) ? v : NEG_SLOPE * v;
  const float p = __expf(v - m[dst * HEADS + hd]);
  atomicAdd(&s[dst * HEADS + hd], p);
}

// ---- pass 3: alpha + scatter out[dst] += h[src] * alpha ----
// 64 threads per edge: hd = t/16, c = t%16.
__global__ void k_edge_scatter(const long long* __restrict__ ei,
                               const float* __restrict__ a_src,
                               const float* __restrict__ a_dst,
                               const float* __restrict__ m,
                               const float* __restrict__ s,
                               const float* __restrict__ hbuf,
                               float* __restrict__ out,
                               float* __restrict__ alpha_out,
                               long long E, long long ET) {
  long long idx = (long long)blockIdx.x * blockDim.x + threadIdx.x;
  if (idx >= ET * HC) return;
  const long long e = idx >> 6;
  const int t = (int)(idx & 63);
  const int hd = t >> 4, c = t & 15;
  long long src, dst;
  edge_pair(ei, e, E, src, dst);
  float v = a_src[src * HEADS + hd] + a_dst[dst * HEADS + hd];
  v = (v > 0.0f) ? v : NEG_SLOPE * v;
  const float alpha = __expf(v - m[dst * HEADS + hd]) /
                      (s[dst * HEADS + hd] + 1e-16f);
  if (c == 0) alpha_out[e * HEADS + hd] = alpha;
  const float hv = hbuf[src * HC + hd * COUT + c];
  atomicAdd(&out[dst * HC + hd * COUT + c], hv * alpha);
}

extern "C" void kernel_launch(void* const* d_in, const int* in_sizes, int n_in,
                              void* d_out, int out_size, void* d_ws, size_t ws_size,
                              hipStream_t stream) {
  const float*     x       = (const float*)d_in[0];
  const long long* ei      = (const long long*)d_in[1];   // int64 [2, E]
  const float*     W       = (const float*)d_in[2];
  const float*     att_src = (const float*)d_in[3];
  const float*     att_dst = (const float*)d_in[4];
  const float*     bias    = (const float*)d_in[5];

  const int       N  = in_sizes[0] / F_IN;
  const long long E  = in_sizes[1] / 2;
  const long long ET = E + N;        // edges + self-loops

  // workspace layout (floats): h | a_src | a_dst | m | s  (~16 MB)
  float* ws    = (float*)d_ws;
  float* hbuf  = ws;
  float* a_src = hbuf  + (size_t)N * HC;
  float* a_dst = a_src + (size_t)N * HEADS;
  float* m     = a_dst + (size_t)N * HEADS;
  float* s     = m     + (size_t)N * HEADS;

  float* out       = (float*)d_out;            // [N, 64]
  float* alpha_out = out + (size_t)N * HC;     // [E+N, 4]

  const int T = 256;

  k_init<<<((size_t)N * HC + T - 1) / T, T, 0, stream>>>(out, bias, m, s, N);

  k_gemm_wmma<<<(N + 15) / 16, 128, 0, stream>>>(x, W, hbuf, N);

  k_attn<<<((size_t)N * HEADS + T - 1) / T, T, 0, stream>>>(
      hbuf, att_src, att_dst, a_src, a_dst, N);

  const long long nEH = ET * HEADS;
  k_edge_max<<<(nEH + T - 1) / T, T, 0, stream>>>(ei, a_src, a_dst, m, E, ET);
  k_edge_sum<<<(nEH + T - 1) / T, T, 0, stream>>>(ei, a_src, a_dst, m, s, E, ET);

  const long long nSC = ET * HC;
  k_edge_scatter<<<(nSC + T - 1) / T, T, 0, stream>>>(
      ei, a_src, a_dst, m, s, hbuf, out, alpha_out, E, ET);
}